// GraphTransformerEncoder_23605140258843
// MI455X (gfx1250) — compile-verified
//
#include <hip/hip_runtime.h>

#define HID 128
#define NHEAD 4
#define DHEAD 128
#define HD 512      // NHEAD*DHEAD
#define DFF 2048
#define LN_EPS 1e-5f
#define SLOPE 0.2f

typedef __attribute__((ext_vector_type(2))) float v2f;
typedef __attribute__((ext_vector_type(8))) float v8f;

// ---- CDNA5 async global->LDS copy (ASYNCcnt) -------------------------------
__device__ __forceinline__ void async_b128(unsigned lds_off, const float* g) {
  asm volatile("global_load_async_to_lds_b128 %0, %1, off"
               :: "v"(lds_off), "v"(g) : "memory");
}
__device__ __forceinline__ void wait_async() {
  asm volatile("s_wait_asynccnt 0x0" ::: "memory");
}
__device__ __forceinline__ unsigned lds_off_of(const void* p) {
  return (unsigned)(uintptr_t)p;   // low 32 bits of generic LDS addr = LDS offset
}

// One wave computes a 16x16 f32 tile of A(16xK) @ B(KxN-slice) accumulated into c.
// A fragment: element (m,k): lane = m + 16*(k>>1), reg = k&1
// B fragment: element (k,n): lane = n + 16*(k>>1), reg = k&1
// C/D: reg r, lane l -> (m = r + 8*(l>>4), n = l&15)
__device__ __forceinline__ v8f wmma_k(const float* A, int lda, const float* B, int ldb,
                                      int K, v8f c, bool reluA) {
  const int lane = threadIdx.x & 31;
  const int hf = lane >> 4;
  const int mr = lane & 15;
  const float* arow = A + (size_t)mr * lda + 2 * hf;
  const float* bcol = B + mr + (size_t)(2 * hf) * ldb;
  for (int k = 0; k < K; k += 4) {
    v2f a, b;
    float a0 = arow[k];
    float a1 = arow[k + 1];
    if (reluA) { a0 = fmaxf(a0, 0.f); a1 = fmaxf(a1, 0.f); }
    a.x = a0; a.y = a1;
    b.x = bcol[(size_t)k * ldb];
    b.y = bcol[(size_t)k * ldb + ldb];
    c = __builtin_amdgcn_wmma_f32_16x16x4_f32(false, a, false, b, (short)0, c,
                                              false, false);
  }
  return c;
}

// C[M x HD] = A[M x HID] @ W[HID x HD] (+ per-column bias).
// One block per 16-row tile; A tile staged in LDS via async copy; 8 waves cover
// the 32 column tiles (4 each).
__global__ __launch_bounds__(256) void k_gemm_node(
    const float* __restrict__ A, const float* __restrict__ W,
    const float* __restrict__ bias, float* __restrict__ C, int nColTiles) {
  __shared__ __align__(16) float As[16][HID];   // 8 KB
  const int rowT = blockIdx.x;
  const float* gA = A + (size_t)rowT * 16 * HID;
  unsigned lbase = lds_off_of(&As[0][0]);
  for (int ch = threadIdx.x; ch < (16 * HID) / 4; ch += 256)
    async_b128(lbase + ch * 16, gA + ch * 4);
  wait_async();
  __syncthreads();

  const int w = threadIdx.x >> 5;
  const int lane = threadIdx.x & 31;
  const int hf = lane >> 4, mr = lane & 15;
  for (int ct = w; ct < nColTiles; ct += 8) {
    int colBase = ct * 16;
    float binit = bias ? bias[colBase + mr] : 0.f;
    v8f c;
    for (int r = 0; r < 8; ++r) c[r] = binit;
    c = wmma_k(&As[0][0], HID, W + colBase, HD, HID, c, false);
    float* Crow = C + (size_t)rowT * 16 * HD + colBase;
    for (int r = 0; r < 8; ++r) Crow[(size_t)(r + 8 * hf) * HD + mr] = c[r];
  }
}

// el[n,h] = <ft[n,h,:], attn_l[h,:]>, er likewise. Block = 1 node, wave = 1 head.
__global__ __launch_bounds__(128) void k_attn_lr(
    const float* __restrict__ ft, const float* __restrict__ al,
    const float* __restrict__ ar, float* __restrict__ el, float* __restrict__ er) {
  int n = blockIdx.x;
  int h = threadIdx.x >> 5;
  int lane = threadIdx.x & 31;
  const float* f = ft + (size_t)n * HD + h * DHEAD;
  const float* wl = al + h * DHEAD;
  const float* wr = ar + h * DHEAD;
  float sl = 0.f, sr = 0.f;
  for (int i = lane; i < DHEAD; i += 32) {
    float v = f[i];
    sl += v * wl[i];
    sr += v * wr[i];
  }
  for (int off = 16; off; off >>= 1) {
    sl += __shfl_xor(sl, off);
    sr += __shfl_xor(sr, off);
  }
  if (lane == 0) { el[n * NHEAD + h] = sl; er[n * NHEAD + h] = sr; }
}

__global__ __launch_bounds__(256) void k_zero(float* __restrict__ p, int n) {
  int i = blockIdx.x * blockDim.x + threadIdx.x;
  if (i < n) p[i] = 0.f;
}

// Edge pass 1: w = exp(leaky_relu(el[src]+er[dst]+e*ce)), z[dst] += w
__global__ __launch_bounds__(256) void k_edge_w(
    const float* __restrict__ ef, const int* __restrict__ src,
    const int* __restrict__ dst, const float* __restrict__ el,
    const float* __restrict__ er, const float* __restrict__ Wfe,
    const float* __restrict__ ae, float* __restrict__ wbuf,
    float* __restrict__ z, int E) {
  __shared__ float ce[NHEAD];
  if (threadIdx.x < NHEAD) {
    float s = 0.f;
    const float* wf = Wfe + threadIdx.x * DHEAD;
    const float* av = ae + threadIdx.x * DHEAD;
    for (int d = 0; d < DHEAD; ++d) s += wf[d] * av[d];
    ce[threadIdx.x] = s;
  }
  __syncthreads();
  int e = blockIdx.x * blockDim.x + threadIdx.x;
  if (e >= E) return;
  int s = src[e], d = dst[e];
  float ev = ef[e];
  for (int h = 0; h < NHEAD; ++h) {
    float sc = el[s * NHEAD + h] + er[d * NHEAD + h] + ev * ce[h];
    sc = sc > 0.f ? sc : SLOPE * sc;
    float w = expf(sc);
    wbuf[e * NHEAD + h] = w;
    atomicAdd(&z[d * NHEAD + h], w);
  }
}

// Edge pass 2: rst[dst] += alpha * ft[src]. One wave per edge, 16 elems/lane.
__global__ __launch_bounds__(256) void k_edge_scatter(
    const int* __restrict__ src, const int* __restrict__ dst,
    const float* __restrict__ ft, const float* __restrict__ wbuf,
    const float* __restrict__ z, float* __restrict__ rst) {
  int e = (blockIdx.x * blockDim.x + threadIdx.x) >> 5;
  int lane = threadIdx.x & 31;
  int s = src[e], d = dst[e];
  float al[NHEAD];
  for (int h = 0; h < NHEAD; ++h) {
    float zz = z[d * NHEAD + h];
    al[h] = wbuf[e * NHEAD + h] / (zz == 0.f ? 1.f : zz);
  }
  const float* fs = ft + (size_t)s * HD;
  float* rd = rst + (size_t)d * HD;
  for (int i = 0; i < 16; ++i) {
    int idx = lane + 32 * i;
    atomicAdd(&rd[idx], fs[idx] * al[idx >> 7]);
  }
}

// x = LN(h0 + relu(relu(rst)@W_mha + b_mha)). Block = 16 nodes, 8 waves.
// rst tile (16x512, 32KB) async-staged in LDS and shared by all 8 waves.
__global__ __launch_bounds__(256) void k_mha_ln1(
    const float* __restrict__ rst, const float* __restrict__ Wm,
    const float* __restrict__ bm, const float* __restrict__ h0,
    const float* __restrict__ g1, const float* __restrict__ bn1,
    float* __restrict__ xout) {
  __shared__ __align__(16) float Rs[16][HD];    // 32 KB
  __shared__ float xs[16][HID];                 // 8 KB
  int rowBase = blockIdx.x * 16;
  const float* gR = rst + (size_t)rowBase * HD;
  unsigned lbase = lds_off_of(&Rs[0][0]);
  for (int ch = threadIdx.x; ch < (16 * HD) / 4; ch += 256)
    async_b128(lbase + ch * 16, gR + ch * 4);
  wait_async();
  __syncthreads();

  int w = threadIdx.x >> 5;
  int lane = threadIdx.x & 31;
  int hf = lane >> 4, mr = lane & 15;
  v8f c = {};
  c = wmma_k(&Rs[0][0], HD, Wm + w * 16, HID, HD, c, true);
  for (int r = 0; r < 8; ++r) xs[r + 8 * hf][w * 16 + mr] = c[r];
  __syncthreads();
  for (int rr = 2 * w; rr < 2 * w + 2; ++rr) {
    int n = rowBase + rr;
    float v[4], s = 0.f, s2 = 0.f;
    for (int j = 0; j < 4; ++j) {
      int col = lane * 4 + j;
      float t = fmaxf(xs[rr][col] + bm[col], 0.f) + h0[(size_t)n * HID + col];
      v[j] = t; s += t; s2 += t * t;
    }
    for (int off = 16; off; off >>= 1) {
      s += __shfl_xor(s, off);
      s2 += __shfl_xor(s2, off);
    }
    float mu = s * (1.f / HID);
    float var = s2 * (1.f / HID) - mu * mu;
    float rs = rsqrtf(var + LN_EPS);
    for (int j = 0; j < 4; ++j) {
      int col = lane * 4 + j;
      xout[(size_t)n * HID + col] = (v[j] - mu) * rs * g1[col] + bn1[col];
    }
  }
}

// out = LN(x + relu(x@W1+b1)@W2 + b2). Block = 16 nodes; the 16x2048 intermediate
// is processed in two 64KB LDS halves; second GEMM accumulates across halves.
// x tile async-staged in LDS (total 72KB LDS, fine on 320KB WGP).
__global__ __launch_bounds__(256) void k_ffn_ln2(
    const float* __restrict__ x, const float* __restrict__ W1,
    const float* __restrict__ b1, const float* __restrict__ W2,
    const float* __restrict__ b2, const float* __restrict__ g2,
    const float* __restrict__ bn2, float* __restrict__ out) {
  __shared__ __align__(16) float t[16][DFF / 2];   // 64 KB
  __shared__ __align__(16) float xsh[16][HID];     // 8 KB
  int rowBase = blockIdx.x * 16;
  const float* gx = x + (size_t)rowBase * HID;
  unsigned lbase = lds_off_of(&xsh[0][0]);
  for (int ch = threadIdx.x; ch < (16 * HID) / 4; ch += 256)
    async_b128(lbase + ch * 16, gx + ch * 4);
  wait_async();
  __syncthreads();

  int w = threadIdx.x >> 5;
  int lane = threadIdx.x & 31;
  int hf = lane >> 4, mr = lane & 15;
  v8f y = {};
  for (int half = 0; half < 2; ++half) {
    // stage 1: t = relu(xsh @ W1[:, half] + b1[half]); 64 col tiles over 8 waves
    for (int ct = w; ct < 64; ct += 8) {
      int gc = half * (DFF / 2) + ct * 16;
      v8f c = {};
      c = wmma_k(&xsh[0][0], HID, W1 + gc, DFF, HID, c, false);
      for (int r = 0; r < 8; ++r)
        t[r + 8 * hf][ct * 16 + mr] = fmaxf(c[r] + b1[gc + mr], 0.f);
    }
    __syncthreads();
    // stage 2: y += t @ W2[half]; wave w owns output cols [16w,16w+16)
    y = wmma_k(&t[0][0], DFF / 2,
               W2 + (size_t)(half * (DFF / 2)) * HID + w * 16, HID,
               DFF / 2, y, false);
    __syncthreads();
  }
  // residual + b2 staged back into (reused) LDS
  for (int r = 0; r < 8; ++r) {
    int row = r + 8 * hf, col = w * 16 + mr;
    t[row][col] = y[r] + b2[col] + xsh[row][col];
  }
  __syncthreads();
  for (int rr = 2 * w; rr < 2 * w + 2; ++rr) {
    int n = rowBase + rr;
    float v[4], s = 0.f, s2 = 0.f;
    for (int j = 0; j < 4; ++j) {
      int col = lane * 4 + j;
      float tv = t[rr][col];
      v[j] = tv; s += tv; s2 += tv * tv;
    }
    for (int off = 16; off; off >>= 1) {
      s += __shfl_xor(s, off);
      s2 += __shfl_xor(s2, off);
    }
    float mu = s * (1.f / HID);
    float var = s2 * (1.f / HID) - mu * mu;
    float rs = rsqrtf(var + LN_EPS);
    for (int j = 0; j < 4; ++j) {
      int col = lane * 4 + j;
      out[(size_t)n * HID + col] = (v[j] - mu) * rs * g2[col] + bn2[col];
    }
  }
}

extern "C" void kernel_launch(void* const* d_in, const int* in_sizes, int n_in,
                              void* d_out, int out_size, void* d_ws, size_t ws_size,
                              hipStream_t stream) {
  const float* h    = (const float*)d_in[0];
  const float* ef   = (const float*)d_in[1];
  const int*   src  = (const int*)d_in[2];
  const int*   dst  = (const int*)d_in[3];
  // d_in[4] = batch_size (only affects a reshape; output is flat identical)
  const float* Wfc  = (const float*)d_in[5];
  const float* al   = (const float*)d_in[6];
  const float* ar   = (const float*)d_in[7];
  const float* Wfe  = (const float*)d_in[8];
  const float* ae   = (const float*)d_in[9];
  const float* Wres = (const float*)d_in[10];
  const float* gb   = (const float*)d_in[11];
  const float* Wm   = (const float*)d_in[12];
  const float* bm   = (const float*)d_in[13];
  const float* g1   = (const float*)d_in[14];
  const float* bn1  = (const float*)d_in[15];
  const float* g2   = (const float*)d_in[16];
  const float* bn2  = (const float*)d_in[17];
  const float* W1   = (const float*)d_in[18];
  const float* b1   = (const float*)d_in[19];
  const float* W2   = (const float*)d_in[20];
  const float* b2   = (const float*)d_in[21];

  const int N = in_sizes[0] / HID;
  const int E = in_sizes[2];

  float* ft  = (float*)d_ws;                    // N*HD
  float* rst = ft + (size_t)N * HD;             // N*HD
  float* el  = rst + (size_t)N * HD;            // N*NHEAD
  float* er  = el + (size_t)N * NHEAD;          // N*NHEAD
  float* z   = er + (size_t)N * NHEAD;          // N*NHEAD
  float* wb  = z + (size_t)N * NHEAD;           // E*NHEAD
  float* xb  = wb + (size_t)E * NHEAD;          // N*HID

  const int nColTiles = HD / 16;

  k_gemm_node<<<N / 16, 256, 0, stream>>>(h, Wfc, nullptr, ft, nColTiles);
  k_gemm_node<<<N / 16, 256, 0, stream>>>(h, Wres, gb, rst, nColTiles);
  k_attn_lr<<<N, 128, 0, stream>>>(ft, al, ar, el, er);
  k_zero<<<(N * NHEAD + 255) / 256, 256, 0, stream>>>(z, N * NHEAD);
  k_edge_w<<<(E + 255) / 256, 256, 0, stream>>>(ef, src, dst, el, er, Wfe, ae, wb, z, E);
  k_edge_scatter<<<(E * 32) / 256, 256, 0, stream>>>(src, dst, ft, wb, z, rst);
  k_mha_ln1<<<N / 16, 256, 0, stream>>>(rst, Wm, bm, h, g1, bn1, xb);
  k_ffn_ln2<<<N / 16, 256, 0, stream>>>(xb, W1, b1, W2, b2, g2, bn2, (float*)d_out);
}